// PatternDetector_23957327577719
// MI455X (gfx1250) — compile-verified
//
#include <hip/hip_runtime.h>
#include <hip/hip_bf16.h>
#include <climits>

// Problem geometry (fixed by the reference): B=131072 rows, L=256 int32 / row.
#define LLEN      256
#define BLOCK     128                 // threads per block = rows per block (4 waves, wave32)
#define TILE_L    32                  // columns per TDM tile
#define NTILES    (LLEN / TILE_L)     // 8 tiles per row
#define ROWSTRIDE 36                  // dwords: 32 data + 4-dword TDM pad -> 16B-aligned rows

typedef __attribute__((ext_vector_type(4))) unsigned int v4u_t;
typedef __attribute__((ext_vector_type(8))) int          v8i_t;
typedef __attribute__((ext_vector_type(4))) int          v4i_t;

#if defined(__gfx1250__) && __has_builtin(__builtin_amdgcn_tensor_load_to_lds)
#define USE_TDM 1
#else
#define USE_TDM 0
#endif

#if USE_TDM
// One Tensor-Data-Mover load of a (BLOCK rows x TILE_L cols) int32 tile into LDS.
// 2D descriptor; pad 4 dwords after every 32 dwords stored -> LDS row stride 36
// dwords (144 B): rows are 16B-aligned so the scan can use ds_load_b128.
// 6-arg builtin form on this toolchain:
//   (uint32x4 g0, int32x8 g1, int32x4 g2, int32x4 g3, int32x8 extra, i32 cpol)
__device__ __forceinline__ void tdm_load_tile(unsigned lds_addr, unsigned long long gaddr)
{
    v4u_t g0;
    g0[0] = 1u;                                          // count=1 (valid), user mode
    g0[1] = lds_addr;                                    // LDS byte address
    g0[2] = (unsigned)(gaddr & 0xffffffffull);           // global_addr[31:0]
    g0[3] = (unsigned)((gaddr >> 32) & 0x01ffffffull)    // global_addr[56:32]
          | (2u << 30);                                  // type = 2 ("image")

    v8i_t g1;
    g1[0] = (2 << 16)        // data_size = 2 -> 4 bytes
          | (1 << 20)        // pad_enable
          | (4 << 22)        // pad_interval code 4 -> every 32 dwords
          | (3 << 25);       // pad_amount  code 3 -> 4 dwords
    g1[1] = (TILE_L & 0xffff) << 16;                       // tensor_dim0[15:0] (=tile; in-bounds)
    g1[2] = ((TILE_L >> 16) & 0xffff)                      // tensor_dim0[31:16]
          | ((BLOCK & 0xffff) << 16);                      // tensor_dim1[15:0]
    g1[3] = ((BLOCK >> 16) & 0xffff)                       // tensor_dim1[31:16]
          | ((TILE_L & 0xffff) << 16);                     // tile_dim0 = 32
    g1[4] = (BLOCK & 0xffff);                              // tile_dim1 = 128, tile_dim2 = 0 (2D)
    g1[5] = LLEN;                                          // tensor_dim0_stride = 256 elems
    g1[6] = 0;
    g1[7] = 0;

    v4i_t z4; z4[0] = 0; z4[1] = 0; z4[2] = 0; z4[3] = 0;  // groups 2/3 unused (<=2D)
    v8i_t z8;
    z8[0] = 0; z8[1] = 0; z8[2] = 0; z8[3] = 0;
    z8[4] = 0; z8[5] = 0; z8[6] = 0; z8[7] = 0;
    __builtin_amdgcn_tensor_load_to_lds(g0, g1, z4, z4, z8, 0);
}
#endif

// Streaming update over the compacted-nonzero subsequence.
//
// Invariants that make this minimal:
//  * prev/prev2 are INT_MAX until the 1st/2nd nonzero, and nonzero afterwards.
//  * v==prev, v==prev2 are self-masking: false when v==0 (prev,prev2 != 0) and
//    false before enough nonzeros exist (INT_MAX never matches v in [1,VOCAB)).
//  * v>prev is false for v==0 (prev >= 1) and false vs the INT_MAX sentinel.
//  * dec is NOT counted: every consecutive-nonzero pair is exactly one of
//    {==, >, <} and there are n-1 pairs, so dec = (n-1) - rep - inc.
__device__ __forceinline__ void pd_update(int v, int& n, int& rep, int& inc,
                                          int& per, int& prev, int& prev2)
{
    rep += (v == prev);
    inc += (v >  prev);
    per += (v == prev2);
    const bool nz = (v != 0);
    prev2 = nz ? prev : prev2;
    prev  = nz ? v    : prev;
    n    += nz;
}

__global__ __launch_bounds__(BLOCK)
void PatternDetector_23957327577719_kernel(const int* __restrict__ x,
                                           float* __restrict__ out)
{
    const int tid = threadIdx.x;
    const long long row0 = (long long)blockIdx.x * BLOCK;

    int n = 0, rep = 0, inc = 0, per = 0;
    int prev = INT_MAX, prev2 = INT_MAX;

#if USE_TDM
    __shared__ __align__(16) int smem[2][BLOCK * ROWSTRIDE];   // 2 x 18 KB, double buffered
    const unsigned lds0 = (unsigned)(size_t)(&smem[0][0]);     // low 32b of generic ptr = LDS offset
    const unsigned lds1 = (unsigned)(size_t)(&smem[1][0]);

    // Scalar (wave-uniform) predicate so exactly wave 0 issues TDM ops
    // (TDM ignores EXEC, so an exec-masked branch would not be safe).
    const bool lead = ((__builtin_amdgcn_readfirstlane(tid) >> 5) == 0);

    const unsigned long long g_row0 =
        (unsigned long long)(size_t)x + (unsigned long long)row0 * (LLEN * 4ull);

    if (lead) tdm_load_tile(lds0, g_row0);                 // prologue: tile 0 -> buf0

    for (int t = 0; t < NTILES; ++t) {
        if (lead) {
            if (t + 1 < NTILES) {
                // prefetch tile t+1 into the other buffer, then wait for tile t
                tdm_load_tile(((t + 1) & 1) ? lds1 : lds0,
                              g_row0 + (unsigned long long)(t + 1) * (TILE_L * 4ull));
                __builtin_amdgcn_s_wait_tensorcnt(1);      // oldest (tile t) complete
            } else {
                __builtin_amdgcn_s_wait_tensorcnt(0);      // last tile complete
            }
        }
        __syncthreads();                                   // publish tile t to all waves

        // Row base = 144*tid bytes (16B aligned) -> ds_load_b128, 8 loads per tile.
        const int4* rowp =
            reinterpret_cast<const int4*>(&smem[t & 1][tid * ROWSTRIDE]);
#pragma unroll
        for (int j = 0; j < TILE_L / 4; ++j) {
            const int4 v = rowp[j];
            pd_update(v.x, n, rep, inc, per, prev, prev2);
            pd_update(v.y, n, rep, inc, per, prev, prev2);
            pd_update(v.z, n, rep, inc, per, prev, prev2);
            pd_update(v.w, n, rep, inc, per, prev, prev2);
        }
        __syncthreads();                                   // buffer may be overwritten next iter
    }
#else
    // Fallback (toolchains without the TDM builtin): direct row reads.
    const int* rp = x + (size_t)(row0 + tid) * LLEN;
#pragma unroll 4
    for (int c = 0; c < LLEN; c += 4) {
        const int4 v = *reinterpret_cast<const int4*>(rp + c);
        pd_update(v.x, n, rep, inc, per, prev, prev2);
        pd_update(v.y, n, rep, inc, per, prev, prev2);
        pd_update(v.z, n, rep, inc, per, prev, prev2);
        pd_update(v.w, n, rep, inc, per, prev, prev2);
    }
#endif

    const int npairs = (n > 0) ? (n - 1) : 0;
    const int dec    = npairs - rep - inc;                 // closed form, see pd_update

    const int   d1  = (n > 2) ? (n - 1) : 1;               // max(n-1, 1)
    const float fd1 = (float)d1;
    float4 f;
    f.x = (float)rep / fd1;
    f.y = (float)inc / fd1;
    f.z = (float)dec / fd1;
    f.w = (n >= 4) ? (float)per / (float)(n - 2) : 0.0f;   // n>=4 -> denom2 = n-2
    if (n <= 1) { f.x = 0.0f; f.y = 0.0f; f.z = 0.0f; f.w = 0.0f; }

    reinterpret_cast<float4*>(out)[row0 + tid] = f;        // coalesced B128 store
}

extern "C" void kernel_launch(void* const* d_in, const int* in_sizes, int n_in,
                              void* d_out, int out_size, void* d_ws, size_t ws_size,
                              hipStream_t stream) {
    (void)n_in; (void)out_size; (void)d_ws; (void)ws_size;
    const int* x   = (const int*)d_in[0];
    float*     out = (float*)d_out;
    const int nrows  = in_sizes[0] / LLEN;   // 131072
    const int blocks = nrows / BLOCK;        // 1024 (exact: 131072 % 128 == 0)
    PatternDetector_23957327577719_kernel<<<dim3(blocks), dim3(BLOCK), 0, stream>>>(x, out);
}